// Similarity_Loss_grouped_SNN_31490700214356
// MI455X (gfx1250) — compile-verified
//
#include <hip/hip_runtime.h>

// ---------------------------------------------------------------------------
// Grouped SNN similarity loss, fused:  d2 -> dist -> exp -> masked row sums.
// GEMM tiles via V_WMMA_F32_16X16X4_F32 (full fp32 precision; GEMM flops are
// tiny, transcendental epilogue dominates, so no reason to downcast).
// ---------------------------------------------------------------------------

typedef __attribute__((ext_vector_type(2))) float v2f;
typedef __attribute__((ext_vector_type(8))) float v8f;

#define BSZ 8192
#define DD 64
#define NCHUNK 8            // column chunks per row tile -> 512*8 = 4096 waves
#define CTILES (512 / NCHUNK)
#define INV_TEMP 20.0f      // 1 / 0.05

// --- kernel 1: row norms + zero accumulators (ws is poisoned by harness) ---
__global__ __launch_bounds__(256) void snn_init(
    const float* __restrict__ text, const float* __restrict__ image,
    float* __restrict__ A, float* __restrict__ den, float* __restrict__ diag,
    float* __restrict__ nImg, float* __restrict__ nTxt) {
  int i = blockIdx.x * blockDim.x + threadIdx.x;
  if (i >= BSZ) return;
  float si = 0.f, st = 0.f;
#pragma unroll 8
  for (int k = 0; k < DD; ++k) {
    float a = image[i * DD + k]; si += a * a;
    float b = text[i * DD + k];  st += b * b;
  }
  nImg[i] = si; nTxt[i] = st;
  A[i] = 0.f; den[i] = 0.f; diag[i] = 0.f;
}

// --- kernel 2: tiled WMMA GEMM + fused exp/mask epilogue + row reduction ---
__global__ __launch_bounds__(256) void snn_tiles(
    const float* __restrict__ text, const float* __restrict__ image,
    const int* __restrict__ groups,
    const float* __restrict__ nImg, const float* __restrict__ nTxt,
    float* __restrict__ A, float* __restrict__ den, float* __restrict__ diag) {
  const int lane = threadIdx.x & 31;
  const int w = blockIdx.x * 8 + (threadIdx.x >> 5);  // global wave id, 0..4095
  const int rt = w >> 3;                              // row tile 0..511
  const int chunk = w & (NCHUNK - 1);                 // column chunk 0..7
  const int i0 = rt * 16;
  const int hi = lane >> 4;                           // half-wave select
  const int lo = lane & 15;
  const int kb = hi * 2;                              // K-pair offset in frag

  // A fragments: 16x4 f32 layout = lane M = lo, VGPR pair K = {4kk+kb, +1}.
  // Whole 16x64 image row-tile cached in 32 VGPRs/lane, reused for 64 tiles.
  v2f a[16];
  {
    const float* arow = image + (size_t)(i0 + lo) * DD + kb;
#pragma unroll
    for (int kk = 0; kk < 16; ++kk)
      a[kk] = *(const v2f*)(arow + kk * 4);
  }

  // Row-side epilogue constants: C layout -> this lane owns rows r + 8*hi.
  float nI[8]; int gR[8];
#pragma unroll
  for (int r = 0; r < 8; ++r) {
    const int i = i0 + r + 8 * hi;
    nI[r] = nImg[i];
    gR[r] = groups[i];
  }

  float accA[8] = {0, 0, 0, 0, 0, 0, 0, 0};
  float accD[8] = {0, 0, 0, 0, 0, 0, 0, 0};

  for (int t = 0; t < CTILES; ++t) {
    const int j0 = (chunk * CTILES + t) * 16;

    // B fragments: 4x16 f32, lane N = lo, K pair = {4kk+kb, +1}.
    v2f b[16];
    const float* brow = text + (size_t)(j0 + lo) * DD + kb;
#pragma unroll
    for (int kk = 0; kk < 16; ++kk)
      b[kk] = *(const v2f*)(brow + kk * 4);

    v8f c = {0.f, 0.f, 0.f, 0.f, 0.f, 0.f, 0.f, 0.f};
#pragma unroll
    for (int kk = 0; kk < 16; ++kk)
      c = __builtin_amdgcn_wmma_f32_16x16x4_f32(
          /*neg_a=*/false, a[kk], /*neg_b=*/false, b[kk],
          /*c_mod=*/(short)0, c, /*reuse_a=*/false, /*reuse_b=*/false);

    const int j = j0 + lo;            // this lane's column index
    const float nT = nTxt[j];
    const int gC = groups[j];

#pragma unroll
    for (int r = 0; r < 8; ++r) {
      const int i = i0 + r + 8 * hi;
      float d2 = fmaxf(nI[r] + nT - 2.0f * c[r], 0.f);
      float s = __expf(-__builtin_amdgcn_sqrtf(d2) * INV_TEMP);
      const bool same = (gR[r] == gC);
      const bool dg = (i == j);
      if (dg) diag[i] = s;            // unique writer over the whole grid
      accA[r] += (same && !dg) ? s : 0.f;
      accD[r] += same ? 0.f : s;
    }
  }

  // Reduce each row's partial sums across its 16 owning lanes (per half-wave).
#pragma unroll
  for (int r = 0; r < 8; ++r) {
    float va = accA[r], vd = accD[r];
#pragma unroll
    for (int off = 1; off < 16; off <<= 1) {
      va += __shfl_xor(va, off, 32);
      vd += __shfl_xor(vd, off, 32);
    }
    if (lo == 0) {
      const int i = i0 + r + 8 * hi;
      atomicAdd(&A[i], va);
      atomicAdd(&den[i], vd);
    }
  }
}

// --- kernel 3: reference's per-row scalar logic + final mean --------------
__global__ __launch_bounds__(256) void snn_finalize(
    const float* __restrict__ A, const float* __restrict__ den,
    const float* __restrict__ diag, const int* __restrict__ groups,
    float* __restrict__ out) {
  __shared__ float sred[256];
  const int tid = threadIdx.x;
  const int g0 = groups[0];
  float local = 0.f;
  for (int i = tid; i < BSZ; i += 256) {
    const float a = A[i], dn = den[i], dgv = diag[i];
    const float F = a + dgv;
    const bool cond = (groups[i] == g0) && (i != 0);
    const float num = cond ? a : (a + F);
    const bool valid = (num != 0.f) && (dn != 0.f);
    const float nn = valid ? num : 1.f;
    const float dd = valid ? dn : 1.f;
    local += valid ? -__logf(nn / dd) : 0.f;
  }
  sred[tid] = local;
  __syncthreads();
  for (int s = 128; s > 0; s >>= 1) {
    if (tid < s) sred[tid] += sred[tid + s];
    __syncthreads();
  }
  if (tid == 0) out[0] = sred[0] / (float)BSZ;
}

extern "C" void kernel_launch(void* const* d_in, const int* in_sizes, int n_in,
                              void* d_out, int out_size, void* d_ws, size_t ws_size,
                              hipStream_t stream) {
  const float* text  = (const float*)d_in[0];   // text_embeddings [B,D]
  const float* image = (const float*)d_in[1];   // image_embeddings [B,D]
  const int*   groups = (const int*)d_in[2];    // groups [B]

  float* ws   = (float*)d_ws;                   // 5*B floats = 160 KB scratch
  float* A    = ws;
  float* den  = ws + BSZ;
  float* diag = ws + 2 * BSZ;
  float* nImg = ws + 3 * BSZ;
  float* nTxt = ws + 4 * BSZ;

  snn_init<<<BSZ / 256, 256, 0, stream>>>(text, image, A, den, diag, nImg, nTxt);
  snn_tiles<<<512, 256, 0, stream>>>(text, image, groups, nImg, nTxt, A, den, diag);
  snn_finalize<<<1, 256, 0, stream>>>(A, den, diag, groups, (float*)d_out);
}